// BaseDA_63874753626444
// MI455X (gfx1250) — compile-verified
//
#include <hip/hip_runtime.h>
#include <hip/hip_bf16.h>
#include <math.h>

typedef __attribute__((ext_vector_type(2))) float v2f;
typedef __attribute__((ext_vector_type(8))) float v8f;

#define NN 4096
#define EE 65536
#define FIN 128
#define HH 64
#define CC 16
#define MM (2 * NN)

// ---------------- small utility kernels ----------------

__global__ void zero_kernel(float* __restrict__ p, int n) {
    int i = blockIdx.x * blockDim.x + threadIdx.x;
    if (i < n) p[i] = 0.0f;
}

// C[M,Nc] = A[M,K] @ B[K,Nc], row-major, naive (tiny matmuls)
__global__ void mm_naive(const float* __restrict__ A, const float* __restrict__ B,
                         float* __restrict__ C, int M, int K, int Nc) {
    int idx = blockIdx.x * blockDim.x + threadIdx.x;
    if (idx >= M * Nc) return;
    int r = idx / Nc, c = idx % Nc;
    const float* a = A + (size_t)r * K;
    float s = 0.0f;
    for (int k = 0; k < K; ++k) s += a[k] * B[(size_t)k * Nc + c];
    C[idx] = s;
}

__global__ void deg_kernel(const int* __restrict__ dst, float* __restrict__ deg, int E) {
    int e = blockIdx.x * blockDim.x + threadIdx.x;
    if (e < E) atomicAdd(&deg[dst[e]], 1.0f);
}

__global__ void norm_kernel(float* __restrict__ norm, int n) {
    int i = blockIdx.x * blockDim.x + threadIdx.x;
    if (i < n) norm[i] = rsqrtf(norm[i] + 1.0f);
}

// agg = X * norm[row]  (self term of GCN propagation)
__global__ void selfinit_kernel(const float* __restrict__ X, const float* __restrict__ norm,
                                float* __restrict__ agg, int n) {
    int idx = blockIdx.x * blockDim.x + threadIdx.x;
    if (idx >= n * HH) return;
    int i = idx >> 6;
    agg[idx] = X[idx] * norm[i];
}

// agg[dst] += X[src] * norm[src]; one thread = (edge, 4-feature group)
__global__ void edge_scatter(const int* __restrict__ src, const int* __restrict__ dst,
                             const float* __restrict__ X, const float* __restrict__ norm,
                             float* __restrict__ agg, int E) {
    int idx = blockIdx.x * blockDim.x + threadIdx.x;
    int e = idx >> 4;
    if (e >= E) return;
    int hq = (idx & 15) << 2;
    int s = src[e], d = dst[e];
    float ns = norm[s];
    const float* xs = X + ((size_t)s << 6) + hq;
    float* ad = agg + ((size_t)d << 6) + hq;
    atomicAdd(ad + 0, xs[0] * ns);
    atomicAdd(ad + 1, xs[1] * ns);
    atomicAdd(ad + 2, xs[2] * ns);
    atomicAdd(ad + 3, xs[3] * ns);
}

// out = leaky(agg * norm[row] + bias[col])
__global__ void finalize_kernel(const float* __restrict__ agg, const float* __restrict__ norm,
                                const float* __restrict__ bias, float* __restrict__ out, int n) {
    int idx = blockIdx.x * blockDim.x + threadIdx.x;
    if (idx >= n * HH) return;
    int i = idx >> 6, h = idx & 63;
    float v = agg[idx] * norm[i] + bias[h];
    out[idx] = v >= 0.0f ? v : 0.01f * v;
}

__global__ void sq_kernel(const float* __restrict__ T, float* __restrict__ sq, int rows) {
    int i = blockIdx.x * blockDim.x + threadIdx.x;
    if (i >= rows) return;
    const float* t = T + ((size_t)i << 6);
    float s = 0.0f;
    for (int h = 0; h < HH; ++h) s += t[h] * t[h];
    sq[i] = s;
}

// ---------------- Gram / MMD kernels (WMMA f32 16x16x4) ----------------
// Symmetric Gram: only upper-triangular 64x64 blocks computed; off-diagonal
// block partials weighted 2x (sign(i,j)==sign(j,i), d2 symmetric).
// scal slots: [0]=sum(d2), [1]=bw, [2]=signed kernel sum, [3]=class loss sum

template <int PASS>
__global__ __launch_bounds__(256) void gram_kernel(const float* __restrict__ T,
                                                   const float* __restrict__ sq,
                                                   float* __restrict__ scal) {
    const int I0 = blockIdx.y * 64;
    const int J0 = blockIdx.x * 64;
    if (J0 < I0) return;  // uniform per block: lower triangle handled by symmetry

    __shared__ float As[64 * HH];
    __shared__ float Bs[64 * HH];
    __shared__ float red[256];

    const int tid = threadIdx.x;

    // Stage 64 full rows of each panel into LDS (contiguous: rows are HH=64 floats)
    const float4* Ag = (const float4*)(T + (size_t)I0 * HH);
    const float4* Bg = (const float4*)(T + (size_t)J0 * HH);
    float4* As4 = (float4*)As;
    float4* Bs4 = (float4*)Bs;
    for (int i = tid; i < 64 * HH / 4; i += 256) {
        As4[i] = Ag[i];
        Bs4[i] = Bg[i];
    }
    __syncthreads();

    const int lane = tid & 31;
    const int wave = tid >> 5;
    const int lo = lane & 15;
    const int hi = lane >> 4;

    float nbw[5];  // -log2(e)/(bw*2^q), so kernel = exp2(d2 * nbw[q])
    if constexpr (PASS == 2) {
        float cur = scal[1];
        #pragma unroll
        for (int q = 0; q < 5; ++q) { nbw[q] = -1.44269504088896340736f / cur; cur *= 2.0f; }
    }

    float local = 0.0f;
    // 16 sub-tiles of 16x16 in the 64x64 block; each of 8 waves takes 2.
    for (int t = wave * 2; t < wave * 2 + 2; ++t) {
        const int mi = (t >> 2) << 4;
        const int ni = (t & 3) << 4;
        v8f acc = {};
        // A (16x4, M=lo, K = k + 2*hi + r) and B (4x16, N=lo, K = k + 2*hi + r):
        // both read total[row + lo][k + 2*hi .. +1] since B is total^T.
        const float* ap = As + (mi + lo) * HH + (hi << 1);
        const float* bp = Bs + (ni + lo) * HH + (hi << 1);
        #pragma unroll
        for (int k = 0; k < HH; k += 4) {
            v2f a; a.x = ap[k]; a.y = ap[k + 1];
            v2f b; b.x = bp[k]; b.y = bp[k + 1];
            acc = __builtin_amdgcn_wmma_f32_16x16x4_f32(false, a, false, b,
                                                        (short)0, acc, false, false);
        }
        // C/D layout: element (M = r + 8*hi, N = lo) in acc[r]
        float sj = sq[J0 + ni + lo];
        #pragma unroll
        for (int r = 0; r < 8; ++r) {
            float d2 = sq[I0 + mi + r + (hi << 3)] + sj - 2.0f * acc[r];
            d2 = fmaxf(d2, 0.0f);
            if constexpr (PASS == 1) {
                local += d2;
            } else {
                float kx = 0.0f;
                #pragma unroll
                for (int q = 0; q < 5; ++q) kx += exp2f(d2 * nbw[q]);
                local += kx;
            }
        }
    }

    red[tid] = local;
    __syncthreads();
    for (int s = 128; s > 0; s >>= 1) {
        if (tid < s) red[tid] += red[tid + s];
        __syncthreads();
    }
    if (tid == 0) {
        float wscale = (I0 == J0) ? 1.0f : 2.0f;  // mirror block via symmetry
        if constexpr (PASS == 1) {
            atomicAdd(&scal[0], wscale * red[0]);
        } else {
            float sign = ((I0 < NN) == (J0 < NN)) ? 1.0f : -1.0f;
            atomicAdd(&scal[2], sign * wscale * red[0]);
        }
    }
}

__global__ void bw_finalize_kernel(float* __restrict__ scal) {
    float m = (float)MM;
    // bw = sum(d2)/(m^2-m), then / KERNEL_MUL^(KERNEL_NUM//2) = /4
    scal[1] = (scal[0] / (m * m - m)) * 0.25f;
}

// ---------------- classifier loss ----------------

__global__ __launch_bounds__(256) void class_loss_kernel(const float* __restrict__ Hm,
                                                         const float* __restrict__ fcw,
                                                         const float* __restrict__ fcb,
                                                         const int* __restrict__ labels,
                                                         float* __restrict__ scal) {
    __shared__ float red[256];
    int tid = threadIdx.x;
    int i = blockIdx.x * blockDim.x + tid;  // i < NN (grid sized exactly)
    float logits[CC];
    #pragma unroll
    for (int c = 0; c < CC; ++c) logits[c] = fcb[c];
    const float* h = Hm + ((size_t)i << 6);
    for (int k = 0; k < HH; ++k) {
        float v = h[k];
        const float* w = fcw + (size_t)k * CC;
        #pragma unroll
        for (int c = 0; c < CC; ++c) logits[c] += v * w[c];
    }
    float mx = logits[0];
    #pragma unroll
    for (int c = 1; c < CC; ++c) mx = fmaxf(mx, logits[c]);
    float se = 0.0f;
    #pragma unroll
    for (int c = 0; c < CC; ++c) se += __expf(logits[c] - mx);
    float lse = __logf(se) + mx;
    red[tid] = lse - logits[labels[i]];
    __syncthreads();
    for (int s = 128; s > 0; s >>= 1) {
        if (tid < s) red[tid] += red[tid + s];
        __syncthreads();
    }
    if (tid == 0) atomicAdd(&scal[3], red[0]);
}

__global__ void final_kernel(const float* __restrict__ scal, float* __restrict__ out) {
    float class_loss = scal[3] / (float)NN;
    float domain_loss = scal[2] / ((float)NN * (float)NN);
    out[0] = class_loss + 0.5f * domain_loss;
}

// ---------------- launch ----------------

extern "C" void kernel_launch(void* const* d_in, const int* in_sizes, int n_in,
                              void* d_out, int out_size, void* d_ws, size_t ws_size,
                              hipStream_t stream) {
    const float* fs   = (const float*)d_in[0];
    const float* ft   = (const float*)d_in[1];
    const float* W1   = (const float*)d_in[2];
    const float* b1   = (const float*)d_in[3];
    const float* W2   = (const float*)d_in[4];
    const float* b2   = (const float*)d_in[5];
    const float* fcw  = (const float*)d_in[6];
    const float* fcb  = (const float*)d_in[7];
    const int* labels = (const int*)d_in[8];
    const int* essrc  = (const int*)d_in[9];
    const int* esdst  = (const int*)d_in[10];
    const int* etsrc  = (const int*)d_in[11];
    const int* etdst  = (const int*)d_in[12];
    (void)in_sizes; (void)n_in; (void)out_size; (void)ws_size;

    float* W = (float*)d_ws;
    float* hs   = W;                    // NN*HH  (hs..ht contiguous => "total")
    float* ht   = hs + NN * HH;         // NN*HH
    float* tmpA = ht + NN * HH;         // NN*HH
    float* tmpB = tmpA + NN * HH;       // NN*HH
    float* norm = tmpB + NN * HH;       // NN
    float* sq   = norm + NN;            // MM
    float* scal = sq + MM;              // 4 scalars

    zero_kernel<<<1, 32, 0, stream>>>(scal, 4);

    const int blk = 256;
    const int gNH = (NN * HH) / blk;    // 1024
    const int gE  = (EE + blk - 1) / blk;
    const int gE16 = (EE * 16) / blk;   // 4096
    const int gN  = NN / blk;

    for (int d = 0; d < 2; ++d) {
        const float* X = d ? ft : fs;
        const int* srcp = d ? etsrc : essrc;
        const int* dstp = d ? etdst : esdst;
        float* outH = d ? ht : hs;

        // layer 1: Z = X @ W1
        mm_naive<<<gNH, blk, 0, stream>>>(X, W1, tmpA, NN, FIN, HH);
        // degrees / norm (same graph for both layers)
        zero_kernel<<<gN + 1, blk, 0, stream>>>(norm, NN);
        deg_kernel<<<gE, blk, 0, stream>>>(dstp, norm, EE);
        norm_kernel<<<gN + 1, blk, 0, stream>>>(norm, NN);
        // propagate + leaky
        selfinit_kernel<<<gNH, blk, 0, stream>>>(tmpA, norm, tmpB, NN);
        edge_scatter<<<gE16, blk, 0, stream>>>(srcp, dstp, tmpA, norm, tmpB, EE);
        finalize_kernel<<<gNH, blk, 0, stream>>>(tmpB, norm, b1, tmpA, NN);
        // layer 2
        mm_naive<<<gNH, blk, 0, stream>>>(tmpA, W2, tmpB, NN, HH, HH);
        selfinit_kernel<<<gNH, blk, 0, stream>>>(tmpB, norm, tmpA, NN);
        edge_scatter<<<gE16, blk, 0, stream>>>(srcp, dstp, tmpB, norm, tmpA, EE);
        finalize_kernel<<<gNH, blk, 0, stream>>>(tmpA, norm, b2, outH, NN);
    }

    // squared norms of total = [hs; ht]
    sq_kernel<<<MM / blk, blk, 0, stream>>>(hs, sq, MM);

    // MMD: pass 1 (sum d2 -> bw), pass 2 (signed kernel sum); upper triangle only
    dim3 gg(MM / 64, MM / 64);  // 128 x 128 blocks, lower triangle early-exits
    gram_kernel<1><<<gg, blk, 0, stream>>>(hs, sq, scal);
    bw_finalize_kernel<<<1, 1, 0, stream>>>(scal);
    gram_kernel<2><<<gg, blk, 0, stream>>>(hs, sq, scal);

    // classifier loss on source
    class_loss_kernel<<<NN / blk, blk, 0, stream>>>(hs, fcw, fcb, labels, scal);

    final_kernel<<<1, 1, 0, stream>>>(scal, (float*)d_out);
}